// CorrelationLayer_22600117911677
// MI455X (gfx1250) — compile-verified
//
#include <hip/hip_runtime.h>

// Correlation layer (FlowNet-style), D=20 stride 2, on gfx1250 via
// V_WMMA_F32_16X16X4_F32 (full fp32, exact-precision vs reference).
//
// Per block (b,h): stage A = f1[b,:,h,:] in LDS once; loop 21 dy values,
// restage f2 row (padded to 144 cols), compute 6x4 = 24 16x16 f32 tiles
// of the w1 x w2p correlation GEMM (K=256) spread over 12 waves
// (2 tiles/wave), scatter the 21 even diagonals to the output.
//
// LDS tiles use channel-PAIR-major layout [c/2][w][c&1] so each fragment's
// (c,w),(c+1,w) values are adjacent dwords: A frag = 1 ds_load_b64,
// both B frags = ds_load_2addr_b64 (dword offsets 0 and 32) ->
// 2 DS ops per K-step. Pair-row strides 224/288 dwords are == 32 (mod 64),
// keeping the two 16-lane K-groups on disjoint bank halves (conflict-free).

typedef __attribute__((ext_vector_type(2))) float v2f;
typedef __attribute__((ext_vector_type(8))) float v8f;

#define BATCH 8
#define CCH   256
#define HH    64
#define WW    96
#define ND    21           // displacements per axis
#define APR   224          // A pair-row stride in dwords (2*112; 192 used)
#define BPR   288          // B pair-row stride in dwords (2*144; w2p in [0,144))
#define NT    384          // 12 waves

__global__ __launch_bounds__(NT, 1)
void corr_wmma_f32_kernel(const float* __restrict__ f1,
                          const float* __restrict__ f2,
                          float* __restrict__ out) {
  extern __shared__ float smem[];
  float* __restrict__ Ald = smem;                      // [128][224] dwords
  float* __restrict__ Bld = smem + (CCH / 2) * APR;    // [128][288] dwords

  const int tid  = threadIdx.x;
  const int lane = tid & 31;
  const int wv   = tid >> 5;         // 0..11
  const int ti   = wv >> 1;          // w1 tile 0..5
  const int djb  = (wv & 1) * 2;     // dj base: 0 or 2

  const int h = blockIdx.x;
  const int b = blockIdx.y;

  const int row = lane & 15;         // M (A) / N (B) within tile
  const int khi = lane >> 4;         // 0: K=k,k+1  1: K=k+2,k+3

  const size_t inBase = (size_t)b * (CCH * HH * WW);

  // ---- stage A = f1[b,:,h,:] (pair-major) ----
  {
    const float* src = f1 + inBase + (size_t)h * WW;
    for (int i = tid; i < CCH * (WW / 4); i += NT) {
      const int c = i / (WW / 4);
      const int q = i % (WW / 4);
      float4 v = *(const float4*)(src + (size_t)c * (HH * WW) + q * 4);
      float* d = &Ald[(c >> 1) * APR + (c & 1) + 8 * q];
      d[0] = v.x; d[2] = v.y; d[4] = v.z; d[6] = v.w;
    }
  }
  // ---- zero B pad columns: w2p in [0,20) and [116,144) (contiguous dwords
  //      in pair-major: [0,40) and [232,288) per pair-row; constant over dy) ----
  {
    const float4 z = make_float4(0.f, 0.f, 0.f, 0.f);
    for (int i = tid; i < (CCH / 2) * 24; i += NT) {
      const int p = i / 24;
      const int j = i % 24;
      const int off = (j < 10) ? (j * 4) : (232 + (j - 10) * 4);
      *(float4*)(&Bld[p * BPR + off]) = z;
    }
  }
  __syncthreads();

  for (int dyI = 0; dyI < ND; ++dyI) {
    const int h2 = h + dyI * 2 - 20;
    const size_t outBase =
        (((size_t)b * (ND * ND) + (size_t)dyI * ND) * HH + (size_t)h) * WW;

    if (h2 < 0 || h2 >= HH) {     // window row entirely in zero padding
      for (int i = tid; i < ND * WW; i += NT) {
        const int dxI = i / WW, w = i % WW;
        out[outBase + (size_t)dxI * (HH * WW) + w] = 0.f;
      }
      continue;                    // uniform branch over the block
    }

    __syncthreads();               // prior compute finished reading Bld
    {
      const float* src = f2 + inBase + (size_t)h2 * WW;
      for (int i = tid; i < CCH * (WW / 4); i += NT) {
        const int c = i / (WW / 4);
        const int q = i % (WW / 4);
        float4 v = *(const float4*)(src + (size_t)c * (HH * WW) + q * 4);
        float* d = &Bld[(c >> 1) * BPR + (c & 1) + 40 + 8 * q];  // w2p = w+20
        d[0] = v.x; d[2] = v.y; d[4] = v.z; d[6] = v.w;
      }
      // warm L2 with next dy's f2 row
      const int h2n = h2 + 2;
      if (h2n < HH) {
        const float* nsrc = f2 + inBase + (size_t)h2n * WW;
        for (int i = tid; i < CCH; i += NT)
          __builtin_prefetch(nsrc + (size_t)i * (HH * WW), 0, 0);
      }
    }
    __syncthreads();

    // ---- compute two 16x16 tiles: (ti, djb) and (ti, djb+1), K=256 ----
    v8f acc0 = {0.f, 0.f, 0.f, 0.f, 0.f, 0.f, 0.f, 0.f};
    v8f acc1 = {0.f, 0.f, 0.f, 0.f, 0.f, 0.f, 0.f, 0.f};

    const float* ap = &Ald[khi * APR + 2 * (ti * 16 + row)];
    const float* bp = &Bld[khi * BPR + 2 * ((ti + djb) * 16 + row)];

#pragma unroll 8
    for (int k = 0; k < CCH; k += 4) {
      v2f a  = *(const v2f*)(ap);        // ds_load_b64
      v2f b0 = *(const v2f*)(bp);        // ds_load_2addr_b64 (with b1)
      v2f b1 = *(const v2f*)(bp + 32);
      acc0 = __builtin_amdgcn_wmma_f32_16x16x4_f32(
          false, a, false, b0, (short)0, acc0, false, false);
      acc1 = __builtin_amdgcn_wmma_f32_16x16x4_f32(
          false, a, false, b1, (short)0, acc1, false, false);
      ap += 2 * APR;
      bp += 2 * BPR;
    }

    // ---- scatter the valid even diagonals (dx = diff-20, diff in [0,40]) ----
    const int mBase = khi * 8;          // C/D: VGPR r -> M = r + 8*(lane/16)
    const int N = lane & 15;
#pragma unroll
    for (int r = 0; r < 8; ++r) {
      const int m  = mBase + r;
      const int w1 = ti * 16 + m;
      {
        const int diff = 16 * djb + N - m;      // w2p - w1
        if (diff >= 0 && diff <= 40 && !(diff & 1))
          out[outBase + (size_t)(diff >> 1) * (HH * WW) + w1] =
              acc0[r] * 0.00390625f;
      }
      {
        const int diff = 16 * (djb + 1) + N - m;
        if (diff >= 0 && diff <= 40 && !(diff & 1))
          out[outBase + (size_t)(diff >> 1) * (HH * WW) + w1] =
              acc1[r] * 0.00390625f;
      }
    }
  }
}

extern "C" void kernel_launch(void* const* d_in, const int* in_sizes, int n_in,
                              void* d_out, int out_size, void* d_ws, size_t ws_size,
                              hipStream_t stream) {
  const float* f1 = (const float*)d_in[0];
  const float* f2 = (const float*)d_in[1];
  float* out = (float*)d_out;

  dim3 grid(HH, BATCH);   // one block per (h, b); dy looped inside
  dim3 block(NT);
  const size_t shmem =
      (size_t)((CCH / 2) * APR + (CCH / 2) * BPR) * sizeof(float); // 256 KB
  corr_wmma_f32_kernel<<<grid, block, shmem, stream>>>(f1, f2, out);
}